// GINEncoder_7258494730854
// MI455X (gfx1250) — compile-verified
//
#include <hip/hip_runtime.h>
#include <hip/hip_bf16.h>

#define N_NODES  100000
#define N_EDGES  1600000
#define N_GRAPHS 1024
#define IN_DIM   7
#define HID      64
#define BN_EPS   1e-5f

typedef float v2f __attribute__((ext_vector_type(2)));
typedef float v8f __attribute__((ext_vector_type(8)));
typedef unsigned u32x4 __attribute__((ext_vector_type(4)));
typedef int   i32x4 __attribute__((ext_vector_type(4)));
typedef int   i32x8 __attribute__((ext_vector_type(8)));

#if __has_include(<hip/amd_detail/amd_gfx1250_TDM.h>)
#define TDM_6ARG 1
#endif

// ---------------------------------------------------------------------------
// gfx1250 async/TDM helpers
// ---------------------------------------------------------------------------
__device__ __forceinline__ unsigned lds_off(const void* p) {
    // Generic LDS pointers are aperture[63:32] | byte-offset[31:0].
    return (unsigned)(unsigned long long)p;
}

// Per-lane 16B global -> LDS async copy (ASYNCcnt tracked).
__device__ __forceinline__ void async_ld_b128(unsigned ldsaddr, const void* gaddr) {
    asm volatile("global_load_async_to_lds_b128 %0, %1, off"
                 :: "v"(ldsaddr), "v"(gaddr) : "memory");
}
__device__ __forceinline__ void wait_async0() {
    asm volatile("s_wait_asynccnt 0x0" ::: "memory");
}

// TDM: load a rows x 64 fp32 tile into LDS with hardware row padding of
// 4 DWORDs every 64 DWORDs (=> stride-68 LDS layout). Rows in
// [rows_tensor, rows_tile) are zero-filled via TDM OOB semantics.
__device__ __forceinline__ void tdm_load_w(const float* gsrc, unsigned ldsaddr,
                                           int rows_tensor, int rows_tile) {
#if __has_builtin(__builtin_amdgcn_tensor_load_to_lds)
    unsigned long long ga = (unsigned long long)gsrc;
    u32x4 g0;
    g0[0] = 1u;                                    // count=1 (valid), user mode
    g0[1] = ldsaddr;                               // lds_addr
    g0[2] = (unsigned)(ga & 0xffffffffu);          // global_addr[31:0]
    g0[3] = (unsigned)((ga >> 32) & 0x01ffffffu)   // global_addr[56:32]
          | 0x80000000u;                           // type=2 ("image")
    i32x8 g1;
    g1[0] = (2 << 16)                              // data_size = 4B
          | (1 << 20)                              // pad_enable
          | (5 << 22)                              // pad_interval: 64 DWORDs
          | (3 << 25);                             // pad_amount:   4 DWORDs
    g1[1] = (64 & 0xffff) << 16;                   // tensor_dim0 = 64
    g1[2] = (rows_tensor & 0xffff) << 16;          // tensor_dim1[15:0]
    g1[3] = (64 & 0xffff) << 16;                   // tensor_dim1[31:16]=0 | tile_dim0=64
    g1[4] = (rows_tile & 0xffff);                  // tile_dim1; tile_dim2=0 (2D)
    g1[5] = 64;                                    // tensor_dim0_stride = 64 elems
    g1[6] = 0;
    g1[7] = 0;
    i32x4 z4 = {0, 0, 0, 0};
#ifdef TDM_6ARG
    i32x8 z8 = {0, 0, 0, 0, 0, 0, 0, 0};
    __builtin_amdgcn_tensor_load_to_lds(g0, g1, z4, z4, z8, 0);
#else
    __builtin_amdgcn_tensor_load_to_lds(g0, g1, z4, z4, 0);
#endif
#endif
}

// ---------------------------------------------------------------------------
// Utility kernels
// ---------------------------------------------------------------------------
__global__ void zero_kernel(float* __restrict__ p, long n) {
    long i = blockIdx.x * (long)blockDim.x + threadIdx.x;
    if (i < n) p[i] = 0.0f;
}

__global__ void copy_kernel(const float* __restrict__ src, float* __restrict__ dst, long n) {
    long i = blockIdx.x * (long)blockDim.x + threadIdx.x;
    if (i < n) dst[i] = src[i];
}

// ---------------------------------------------------------------------------
// Edge aggregation: agg[dst] += x[src]  (one thread per (edge, feature))
// Feature table (25.6MB) fits in 192MB L2 -> atomics stay on-chip.
// ---------------------------------------------------------------------------
__global__ void edge_agg_kernel(const float* __restrict__ x,
                                const int* __restrict__ src,
                                const int* __restrict__ dst,
                                float* __restrict__ agg,
                                int dim, long total) {
    long idx = blockIdx.x * (long)blockDim.x + threadIdx.x;
    if (idx >= total) return;
    int e = (int)(idx / dim);
    int f = (int)(idx % dim);
    int s = src[e];
    int d = dst[e];
    float* tgt = &agg[(long)d * dim + f];
    __builtin_prefetch(tgt, 1, 1);   // global_prefetch_b8: warm scatter target in L2
    atomicAdd(tgt, x[(long)s * dim + f]);
}

// ---------------------------------------------------------------------------
// Fused GIN MLP: h2 = (relu(agg @ W1 + b1)) @ W2 + b2   (fp32 WMMA)
// Weights staged by TDM with hardware padding; A-tiles by async LDS DMA.
// Also accumulates per-column sum/sumsq for training-mode BatchNorm.
// 256 threads = 8 waves; each wave computes a 16x64 output tile.
// ---------------------------------------------------------------------------
__global__ __launch_bounds__(256) void mlp_wmma_kernel(
        const float* __restrict__ agg,
        const float* __restrict__ W1, const float* __restrict__ b1,
        const float* __restrict__ W2, const float* __restrict__ b2,
        float* __restrict__ h2, float* __restrict__ gstats,
        int n, int din) {
    // Stride-68 LDS layouts (matches TDM pad: +4 DWORDs per 64 DWORDs).
    __shared__ float sW1[64 * 68];
    __shared__ float sW2[64 * 68];
    __shared__ float sb1[64], sb2[64];
    __shared__ float sA[8][16][68];     // per-wave A tile, reused as h1 tile
    __shared__ float sSum[64], sSumSq[64];

    const int tid  = threadIdx.x;
    const int K1   = (din + 3) & ~3;           // 8 for layer0 (din=7), else 64
    const int ksh  = (K1 == 8) ? 3 : 6;

#if __has_builtin(__builtin_amdgcn_tensor_load_to_lds)
    // Tensor Data Mover: stage W1 (zero-padding rows din..K1-1 via OOB) and
    // W2 (full 64x64) straight into padded LDS. One wave issues both DMAs.
    if (tid < 32) {
        tdm_load_w(W1, lds_off(&sW1[0]), din, K1);
        tdm_load_w(W2, lds_off(&sW2[0]), 64, 64);
        __builtin_amdgcn_s_wait_tensorcnt(0);
    }
#else
    for (int idx = tid; idx < 64 * 64; idx += 256) {
        int k = idx >> 6, c = idx & 63;
        sW1[k * 68 + c] = (k < din) ? W1[k * 64 + c] : 0.0f;
        sW2[k * 68 + c] = W2[idx];
    }
#endif
    if (tid < 64) {
        sb1[tid] = b1[tid];
        sb2[tid] = b2[tid];
        sSum[tid] = 0.0f;
        sSumSq[tid] = 0.0f;
    }

    const int wave = tid >> 5;
    const int lane = tid & 31;
    const int m    = lane & 15;          // row (A) / col (B) within 16
    const int kh   = (lane >> 4) << 1;   // K sub-offset: 0 or 2
    const int rowbase = blockIdx.x * 128 + wave * 16;

    // Stage this wave's 16 x K1 A-tile.
    if (din == 64) {
        // Contiguous 4KB per wave: async global->LDS, VGPR-bypassing.
        // Tail-block overreads stay inside d_ws; garbage rows never stored.
        const float* gbase = agg + (long)rowbase * 64;
#pragma unroll
        for (int it = 0; it < 8; it++) {
            int t = it * 128 + lane * 4;             // element index in tile
            async_ld_b128(lds_off(&sA[wave][t >> 6][t & 63]), gbase + t);
        }
        wait_async0();
    } else {
        // Layer 0: ragged stride-7 rows, guarded scalar staging w/ zero pad.
        for (int idx = lane; idx < (16 << ksh); idx += 32) {
            int r = idx >> ksh;
            int c = idx & (K1 - 1);
            int grow = rowbase + r;
            sA[wave][r][c] = (grow < n && c < din) ? agg[(long)grow * din + c] : 0.0f;
        }
    }
    __syncthreads();

    // ---- GEMM1: acc = A(16xK1) @ W1(K1x64), fp32 WMMA 16x16x4 ----
    v8f acc[4] = {};
    for (int k = 0; k < K1; k += 4) {
        v2f a;
        a.x = sA[wave][m][k + kh];
        a.y = sA[wave][m][k + kh + 1];
#pragma unroll
        for (int nt = 0; nt < 4; nt++) {
            v2f b;
            b.x = sW1[(k + kh) * 68 + nt * 16 + m];
            b.y = sW1[(k + kh + 1) * 68 + nt * 16 + m];
            acc[nt] = __builtin_amdgcn_wmma_f32_16x16x4_f32(
                false, a, false, b, (short)0, acc[nt], false, false);
        }
    }
    __syncthreads();

    // Epilogue 1: bias + ReLU, write h1 back to LDS in row-major (A layout).
    // C/D layout: lane L, vgpr r -> M = r + 8*(L>>4), N = nt*16 + (L&15)
#pragma unroll
    for (int nt = 0; nt < 4; nt++) {
#pragma unroll
        for (int r = 0; r < 8; r++) {
            int M  = r + ((lane >> 4) << 3);
            int nn = nt * 16 + m;
            sA[wave][M][nn] = fmaxf(acc[nt][r] + sb1[nn], 0.0f);
        }
    }
    __syncthreads();

    // ---- GEMM2: acc2 = h1(16x64) @ W2(64x64) ----
    v8f acc2[4] = {};
    for (int k = 0; k < 64; k += 4) {
        v2f a;
        a.x = sA[wave][m][k + kh];
        a.y = sA[wave][m][k + kh + 1];
#pragma unroll
        for (int nt = 0; nt < 4; nt++) {
            v2f b;
            b.x = sW2[(k + kh) * 68 + nt * 16 + m];
            b.y = sW2[(k + kh + 1) * 68 + nt * 16 + m];
            acc2[nt] = __builtin_amdgcn_wmma_f32_16x16x4_f32(
                false, a, false, b, (short)0, acc2[nt], false, false);
        }
    }

    // Epilogue 2: bias, store h2, accumulate BN statistics (per-column).
#pragma unroll
    for (int nt = 0; nt < 4; nt++) {
        int nn = nt * 16 + m;
        float ps = 0.0f, pss = 0.0f;
#pragma unroll
        for (int r = 0; r < 8; r++) {
            int M = r + ((lane >> 4) << 3);
            int grow = rowbase + M;
            float v = acc2[nt][r] + sb2[nn];
            if (grow < n) {
                h2[(long)grow * 64 + nn] = v;
                ps  += v;
                pss += v * v;
            }
        }
        atomicAdd(&sSum[nn], ps);
        atomicAdd(&sSumSq[nn], pss);
    }
    __syncthreads();

    if (tid < 64) {
        atomicAdd(&gstats[tid],      sSum[tid]);
        atomicAdd(&gstats[64 + tid], sSumSq[tid]);
    }
}

// ---------------------------------------------------------------------------
// BatchNorm (training-mode batch stats) + ReLU
// ---------------------------------------------------------------------------
__global__ void bn_relu_kernel(const float* __restrict__ h2,
                               const float* __restrict__ stats,
                               const float* __restrict__ gamma,
                               const float* __restrict__ beta,
                               float* __restrict__ xout, int n) {
    int idx = blockIdx.x * blockDim.x + threadIdx.x;
    if (idx >= n * 64) return;
    int c = idx & 63;
    float invn = 1.0f / (float)n;
    float mean = stats[c] * invn;
    float var  = stats[64 + c] * invn - mean * mean;
    float v = (h2[idx] - mean) * rsqrtf(var + BN_EPS) * gamma[c] + beta[c];
    xout[idx] = fmaxf(v, 0.0f);
}

// ---------------------------------------------------------------------------
// Global mean pool (sums + counts via atomics)
// ---------------------------------------------------------------------------
__global__ void pool_kernel(const float* __restrict__ xf,
                            const int* __restrict__ batch,
                            float* __restrict__ pool,
                            float* __restrict__ cnt, int n) {
    int idx = blockIdx.x * blockDim.x + threadIdx.x;
    if (idx >= n * 64) return;
    int i = idx >> 6, c = idx & 63;
    int g = batch[i];
    atomicAdd(&pool[g * 64 + c], xf[idx]);
    if (c == 0) atomicAdd(&cnt[g], 1.0f);
}

// ---------------------------------------------------------------------------
// Final projection (64x64) + row L2-normalize.  One block (64 thr) per graph.
// ---------------------------------------------------------------------------
__global__ __launch_bounds__(64) void proj_norm_kernel(
        const float* __restrict__ pool, const float* __restrict__ cnt,
        const float* __restrict__ W, const float* __restrict__ b,
        float* __restrict__ out) {
    __shared__ float srow[64];
    __shared__ float sq[64];
    int g = blockIdx.x, t = threadIdx.x;
    float c = fmaxf(cnt[g], 1.0f);
    srow[t] = pool[g * 64 + t] / c;
    __syncthreads();
    float acc = b[t];
#pragma unroll 8
    for (int k = 0; k < 64; k++) acc += srow[k] * W[k * 64 + t];
    sq[t] = acc * acc;
    __syncthreads();
    for (int s = 32; s > 0; s >>= 1) {
        if (t < s) sq[t] += sq[t + s];
        __syncthreads();
    }
    float norm = sqrtf(sq[0]);
    out[g * 64 + t] = acc / fmaxf(norm, 1e-12f);
}

// ---------------------------------------------------------------------------
// Launch
// ---------------------------------------------------------------------------
static inline unsigned gblocks(long n, int bs) { return (unsigned)((n + bs - 1) / bs); }

extern "C" void kernel_launch(void* const* d_in, const int* in_sizes, int n_in,
                              void* d_out, int out_size, void* d_ws, size_t ws_size,
                              hipStream_t stream) {
    const float* x     = (const float*)d_in[0];
    const int*   ei    = (const int*)d_in[1];
    const int*   batch = (const int*)d_in[2];
    const int*   src   = ei;
    const int*   dst   = ei + N_EDGES;

    const float *W1[3], *B1[3], *W2[3], *B2[3], *G[3], *BE[3];
    for (int l = 0; l < 3; l++) {
        int base = 3 + l * 6;
        W1[l] = (const float*)d_in[base + 0];
        B1[l] = (const float*)d_in[base + 1];
        W2[l] = (const float*)d_in[base + 2];
        B2[l] = (const float*)d_in[base + 3];
        G[l]  = (const float*)d_in[base + 4];
        BE[l] = (const float*)d_in[base + 5];
    }
    const float* projW = (const float*)d_in[21];
    const float* projB = (const float*)d_in[22];

    // Workspace layout (floats)
    float* ws    = (float*)d_ws;
    const size_t NF = (size_t)N_NODES * 64;
    float* AGG   = ws;
    float* H2    = AGG + NF;
    float* XA    = H2 + NF;
    float* XB    = XA + NF;
    float* STATS = XB + NF;                 // 128 floats
    float* POOL  = STATS + 128;             // 1024*64
    float* CNT   = POOL + (size_t)N_GRAPHS * 64;  // 1024

    const float* xin = x;
    float* xout = XA;
    for (int l = 0; l < 3; l++) {
        int din = (l == 0) ? IN_DIM : HID;
        zero_kernel<<<1, 128, 0, stream>>>(STATS, 128);
        long cn = (long)N_NODES * din;
        copy_kernel<<<gblocks(cn, 256), 256, 0, stream>>>(xin, AGG, cn);
        long ce = (long)N_EDGES * din;
        edge_agg_kernel<<<gblocks(ce, 256), 256, 0, stream>>>(xin, src, dst, AGG, din, ce);
        mlp_wmma_kernel<<<gblocks(N_NODES, 128), 256, 0, stream>>>(
            AGG, W1[l], B1[l], W2[l], B2[l], H2, STATS, N_NODES, din);
        bn_relu_kernel<<<gblocks((long)N_NODES * 64, 256), 256, 0, stream>>>(
            H2, STATS, G[l], BE[l], xout, N_NODES);
        xin  = xout;
        xout = (l == 0) ? XB : XA;
    }

    long pz = (long)N_GRAPHS * 64 + N_GRAPHS;
    zero_kernel<<<gblocks(pz, 256), 256, 0, stream>>>(POOL, pz);
    pool_kernel<<<gblocks((long)N_NODES * 64, 256), 256, 0, stream>>>(
        xin, batch, POOL, CNT, N_NODES);
    proj_norm_kernel<<<N_GRAPHS, 64, 0, stream>>>(POOL, CNT, projW, projB, (float*)d_out);
}